// Attention_2027224563763
// MI455X (gfx1250) — compile-verified
//
#include <hip/hip_runtime.h>

typedef __attribute__((ext_vector_type(2))) float v2f;
typedef __attribute__((ext_vector_type(8))) float v8f;
typedef int b128v __attribute__((__vector_size__(16)));   // matches builtin param type

#define B_ 256
#define L_ 196
#define E_ 2048
#define A_ 512
#define LP 208          // 13 * 16 padded rows
#define KC 32           // K chunk staged in LDS
#define LDA 44          // lsA row stride (floats): 176B = 16B-multiple, 16 distinct banks
#define LDB 132         // lsB row stride (floats): 528B = 16B-multiple
#define NT 8            // 16-wide n-tiles per group per wave (128 cols)
#define NGROUPS 4       // 4 * 128 = 512 = A_
#define NCHUNK (NGROUPS * (E_ / KC))   // 256 pipelined chunks

#define ASZ (LP * LDA)  // one lsA buffer, floats
#define BSZ (KC * LDB)  // one lsB buffer, floats

#if __has_builtin(__builtin_amdgcn_global_load_async_to_lds_b128)
#define ASYNC_COPY 1
#else
#define ASYNC_COPY 0
#endif

__device__ __forceinline__ void copy16_g2l(const float* __restrict__ g, float* l) {
#if ASYNC_COPY
    __builtin_amdgcn_global_load_async_to_lds_b128((b128v*)g, (b128v*)l, 0, 0);
#else
    *(float4*)l = *(const float4*)g;   // sync fallback
#endif
}

__device__ __forceinline__ void wait_async_tail5() {
#if ASYNC_COPY
#if __has_builtin(__builtin_amdgcn_s_wait_asynccnt)
    __builtin_amdgcn_s_wait_asynccnt(5);
#else
    asm volatile("s_wait_asynccnt 5" ::: "memory");
#endif
#endif
}

__device__ __forceinline__ void wait_async_all() {
#if ASYNC_COPY
#if __has_builtin(__builtin_amdgcn_s_wait_asynccnt)
    __builtin_amdgcn_s_wait_asynccnt(0);
#else
    asm volatile("s_wait_asynccnt 0" ::: "memory");
#endif
#endif
}

// Stage chunk (k0 of img rows, [k0, n0] block of Ww) into buffer `Ab/Bb`.
// Exactly 5 async B128 instructions per thread (wave-uniform), plus a small
// synchronous remainder (img rows 192..195) handled by 128 threads.
__device__ __forceinline__ void stage_chunk(const float* __restrict__ imgb,
                                            const float* __restrict__ Ww,
                                            float* Ab, float* Bb,
                                            int k0, int n0, int t) {
    #pragma unroll
    for (int j = 0; j < 3; ++j) {          // img rows 0..191: 1536 x 16B
        int i = t + j * 512;
        int row = i >> 3, seg = i & 7;
        copy16_g2l(imgb + row * E_ + k0 + seg * 4, Ab + row * LDA + seg * 4);
    }
    #pragma unroll
    for (int j = 0; j < 2; ++j) {          // Ww KC x 128: 1024 x 16B
        int i = t + j * 512;
        int row = i >> 5, seg = i & 31;
        copy16_g2l(Ww + (k0 + row) * A_ + n0 + seg * 4, Bb + row * LDB + seg * 4);
    }
    if (t < 128) {                         // img rows 192..195 remainder (sync)
        int row = 192 + (t >> 5), col = t & 31;
        Ab[row * LDA + col] = imgb[row * E_ + k0 + col];
    }
}

__launch_bounds__(512, 1)
__global__ void att_fused(const float* __restrict__ img,
                          const float* __restrict__ hid,
                          const float* __restrict__ Uw,
                          const float* __restrict__ Ub,
                          const float* __restrict__ Ww,
                          const float* __restrict__ Wb,
                          const float* __restrict__ vw,
                          float* __restrict__ out_ctx,
                          float* __restrict__ out_alpha)
{
    __shared__ float lsA[2 * ASZ];         // double-buffered img K-chunks
    __shared__ float lsB[2 * BSZ];         // double-buffered Ww K-chunks
    __shared__ float lsUH[A_];
    __shared__ float lsE[LP];
    __shared__ float lsAlpha[L_];
    __shared__ float red[16];

    const int b    = blockIdx.x;
    const int t    = threadIdx.x;
    const int wave = t >> 5;
    const int lane = t & 31;
    const int half = lane >> 4;
    const int lr   = lane & 15;

    // ---------- Phase 0: uh[a] = Ub[a] + Wb[a] + sum_k hid[b,k]*Uw[k,a] ----------
    {
        float acc = Ub[t] + Wb[t];
        const float* hb = hid + b * A_;
        for (int k = 0; k < A_; ++k)
            acc = fmaf(hb[k], Uw[k * A_ + t], acc);
        lsUH[t] = acc;
    }
    if (t < LP) lsE[t] = 0.0f;
    // zero the pad rows (196..207, cols 0..31) of both lsA buffers once
    for (int i = t; i < 2 * 12 * 32; i += 512) {
        int bufi = i / 384, r = (i % 384) >> 5, c = i & 31;
        lsA[bufi * ASZ + (196 + r) * LDA + c] = 0.0f;
    }

    const float* imgb = img + (size_t)b * L_ * E_;

    // ---------- Phase 1: e[l] = v . tanh(img@Ww + uh), WMMA + async double-buffer ----------
    v8f acc[NT];
    v8f zero = {};
    #pragma unroll
    for (int i = 0; i < NT; ++i) acc[i] = zero;

    // prologue: start chunk 0 into buffer 0
    stage_chunk(imgb, Ww, lsA, lsB, 0, 0, t);

    for (int c = 0; c < NCHUNK; ++c) {
        const int buf = c & 1;
        const int ng  = c >> 6;                 // 64 chunks per n-group
        float* Ab = lsA + buf * ASZ;
        float* Bb = lsB + buf * BSZ;

        if (c + 1 < NCHUNK) {                   // stream next chunk into other buffer
            int cn  = c + 1;
            int k0n = (cn & 63) * KC;
            int n0n = (cn >> 6) * 128;
            stage_chunk(imgb, Ww, lsA + (buf ^ 1) * ASZ, lsB + (buf ^ 1) * BSZ,
                        k0n, n0n, t);
            wait_async_tail5();                 // only the 5 newest may remain in flight
        } else {
            wait_async_all();
        }
        __syncthreads();                        // chunk c visible to all waves

        if (wave < 13) {
            const int m0 = wave * 16;
            #pragma unroll
            for (int ks = 0; ks < KC / 4; ++ks) {
                // A frag 16x4 f32: lane = M row, half selects K pair
                v2f a = *(const v2f*)&Ab[(m0 + lr) * LDA + ks * 4 + 2 * half];
                #pragma unroll
                for (int nt = 0; nt < NT; ++nt) {
                    v2f bb;
                    bb.x = Bb[(ks * 4 + 2 * half)     * LDB + nt * 16 + lr];
                    bb.y = Bb[(ks * 4 + 2 * half + 1) * LDB + nt * 16 + lr];
                    acc[nt] = __builtin_amdgcn_wmma_f32_16x16x4_f32(
                        false, a, false, bb, (short)0, acc[nt], false, false);
                }
            }
        }

        if ((c & 63) == 63) {                   // n-group complete: tanh + v-dot epilogue
            if (wave < 13) {
                const int m0 = wave * 16;
                const int n0 = ng * 128;
                #pragma unroll
                for (int nt = 0; nt < NT; ++nt) {
                    int col   = n0 + nt * 16 + lr;
                    float uhv = lsUH[col];
                    float vv  = vw[col];
                    #pragma unroll
                    for (int i = 0; i < 8; ++i) {
                        float val = tanhf(acc[nt][i] + uhv) * vv;
                        val += __shfl_xor(val, 1, 32);
                        val += __shfl_xor(val, 2, 32);
                        val += __shfl_xor(val, 4, 32);
                        val += __shfl_xor(val, 8, 32);
                        if (lr == 0) lsE[m0 + i + 8 * half] += val;  // lanes 0 & 16
                    }
                }
            }
            #pragma unroll
            for (int i = 0; i < NT; ++i) acc[i] = zero;
        }
        __syncthreads();                        // buf[c&1] free before it is re-filled
    }

    // ---------- Phase 2: softmax over L ----------
    float x = (t < L_) ? lsE[t] : -__builtin_inff();
    float m = x;
    #pragma unroll
    for (int off = 16; off >= 1; off >>= 1) m = fmaxf(m, __shfl_xor(m, off, 32));
    if (lane == 0) red[wave] = m;
    __syncthreads();
    if (wave == 0) {
        float mm = red[lane & 15];
        #pragma unroll
        for (int off = 8; off >= 1; off >>= 1) mm = fmaxf(mm, __shfl_xor(mm, off, 32));
        if (lane == 0) red[0] = mm;
    }
    __syncthreads();
    float gmax = red[0];
    __syncthreads();
    float ex = (t < L_) ? expf(x - gmax) : 0.0f;
    float s = ex;
    #pragma unroll
    for (int off = 16; off >= 1; off >>= 1) s += __shfl_xor(s, off, 32);
    if (lane == 0) red[wave] = s;
    __syncthreads();
    if (wave == 0) {
        float ss = red[lane & 15];
        #pragma unroll
        for (int off = 8; off >= 1; off >>= 1) ss += __shfl_xor(ss, off, 32);
        if (lane == 0) red[0] = ss;
    }
    __syncthreads();
    float inv = 1.0f / red[0];
    if (t < L_) {
        float alpha = ex * inv;
        lsAlpha[t] = alpha;
        out_alpha[b * L_ + t] = alpha;
    }
    __syncthreads();

    // ---------- Phase 3: context[b,:] = sum_l alpha[l] * img[b,l,:] ----------
    float c0 = 0.f, c1 = 0.f, c2 = 0.f, c3 = 0.f;
    for (int l = 0; l < L_; ++l) {
        float a = lsAlpha[l];
        const float* r = imgb + l * E_;
        c0 = fmaf(a, r[t], c0);
        c1 = fmaf(a, r[t + 512], c1);
        c2 = fmaf(a, r[t + 1024], c2);
        c3 = fmaf(a, r[t + 1536], c3);
    }
    float* oc = out_ctx + (size_t)b * E_;
    oc[t] = c0; oc[t + 512] = c1; oc[t + 1024] = c2; oc[t + 1536] = c3;
}

extern "C" void kernel_launch(void* const* d_in, const int* in_sizes, int n_in,
                              void* d_out, int out_size, void* d_ws, size_t ws_size,
                              hipStream_t stream) {
    const float* img = (const float*)d_in[0];
    const float* hid = (const float*)d_in[1];
    const float* Uw  = (const float*)d_in[2];
    const float* Ub  = (const float*)d_in[3];
    const float* Ww  = (const float*)d_in[4];
    const float* Wb  = (const float*)d_in[5];
    const float* vw  = (const float*)d_in[6];
    // v_b shifts e uniformly -> softmax invariant; not needed.

    float* out_ctx   = (float*)d_out;                 // [256, 2048]
    float* out_alpha = (float*)d_out + B_ * E_;       // [256, 196]

    att_fused<<<B_, 512, 0, stream>>>(img, hid, Uw, Ub, Ww, Wb, vw, out_ctx, out_alpha);
}